// GraphExtractor_55087250538901
// MI455X (gfx1250) — compile-verified
//
#include <hip/hip_runtime.h>

// Problem constants from the reference
#define G_NUM   512
#define NCOLORS 32
#define FIN     64   // == F_OUT

typedef float v2f __attribute__((ext_vector_type(2)));
typedef float v8f __attribute__((ext_vector_type(8)));

// ---------------- workspace zeroing ----------------
__global__ void zero_ws_kernel(float* __restrict__ p, int n) {
    int i = blockIdx.x * blockDim.x + threadIdx.x;
    if (i < n) p[i] = 0.0f;
}

// ---------------- streaming pass over u (HBM-bound) ----------------
// Each 16-thread "column" owns a contiguous chunk of nodes; thread q of a
// column handles feature quad [4q..4q+3]. Register-accumulate per graph
// segment (batch is sorted), flush with hardware FP atomics on segment change.
__device__ __forceinline__ void flush_seg(int g, int q,
                                          const float4& accD, const float4& accM,
                                          float localcnt,
                                          float* __restrict__ sum_diag,
                                          float* __restrict__ sum_mean,
                                          float* __restrict__ cnt) {
    float* pd = sum_diag + g * FIN + q * 4;
    float* pm = sum_mean + g * FIN + q * 4;
    unsafeAtomicAdd(pd + 0, accD.x);
    unsafeAtomicAdd(pd + 1, accD.y);
    unsafeAtomicAdd(pd + 2, accD.z);
    unsafeAtomicAdd(pd + 3, accD.w);
    unsafeAtomicAdd(pm + 0, accM.x);
    unsafeAtomicAdd(pm + 1, accM.y);
    unsafeAtomicAdd(pm + 2, accM.z);
    unsafeAtomicAdd(pm + 3, accM.w);
    if (q == 0) unsafeAtomicAdd(cnt + g, localcnt);
}

__global__ __launch_bounds__(256)
void accum_kernel(const float* __restrict__ u,
                  const int* __restrict__ coloring,
                  const int* __restrict__ batch,
                  const float* __restrict__ n_batch,
                  float* __restrict__ sum_diag,
                  float* __restrict__ sum_mean,
                  float* __restrict__ cnt,
                  int N, int columns) {
    int tid    = blockIdx.x * blockDim.x + threadIdx.x;
    int q      = tid & 15;        // feature quad
    int col_id = tid >> 4;        // node-stream id
    int chunk  = (N + columns - 1) / columns;
    int n0     = col_id * chunk;
    int n1     = n0 + chunk; if (n1 > N) n1 = N;

    const float4* u4 = (const float4*)u;

    float4 accD = {0.f, 0.f, 0.f, 0.f};
    float4 accM = {0.f, 0.f, 0.f, 0.f};
    float  localcnt = 0.0f;
    int    curg = -1;

    for (int n = n0; n < n1; ++n) {
        int   cl  = coloring[n];
        float inv = 1.0f / n_batch[n];
        const float4* up = u4 + (size_t)n * (NCOLORS * FIN / 4) + q;

        float4 s = {0.f, 0.f, 0.f, 0.f};
        float4 d = {0.f, 0.f, 0.f, 0.f};
#pragma unroll
        for (int c = 0; c < NCOLORS; ++c) {
            float4 v = up[c * (FIN / 4)];           // 128-bit coalesced load
            s.x += v.x; s.y += v.y; s.z += v.z; s.w += v.w;
            if (c == cl) d = v;                     // v_cndmask select
        }

        int g = batch[n];
        if (g != curg) {
            if (curg >= 0)
                flush_seg(curg, q, accD, accM, localcnt, sum_diag, sum_mean, cnt);
            curg = g;
            accD = make_float4(0.f, 0.f, 0.f, 0.f);
            accM = make_float4(0.f, 0.f, 0.f, 0.f);
            localcnt = 0.0f;
        }
        accD.x += d.x;       accD.y += d.y;       accD.z += d.z;       accD.w += d.w;
        accM.x += s.x * inv; accM.y += s.y * inv; accM.z += s.z * inv; accM.w += s.w * inv;
        localcnt += 1.0f;
    }
    if (curg >= 0)
        flush_seg(curg, q, accD, accM, localcnt, sum_diag, sum_mean, cnt);
}

// ---------------- turn sums into pooled operands ----------------
// X1 = sum_diag / max(cnt,1)          (mean_batch_trace)
// X2 = sum_mean / max(cnt,1) - X1     (batch_sum)
__global__ void finalize_kernel(const float* __restrict__ sum_diag,
                                const float* __restrict__ sum_mean,
                                const float* __restrict__ cnt,
                                float* __restrict__ X1,
                                float* __restrict__ X2) {
    int tid = blockIdx.x * blockDim.x + threadIdx.x;
    if (tid >= G_NUM * FIN) return;
    int   g   = tid >> 6;
    float inv = 1.0f / fmaxf(cnt[g], 1.0f);
    float x1  = sum_diag[tid] * inv;
    X1[tid] = x1;
    X2[tid] = sum_mean[tid] * inv - x1;
}

// ---------------- WMMA epilogue ----------------
// pre  = X1@W1 + X2@W2 + b1 + b2      (512 x 64)
// out  = pre + relu(pre)@Wlin + blin
// 32 blocks (16-row tiles) x 4 waves (16-col tiles). V_WMMA_F32_16X16X4_F32.
__global__ __launch_bounds__(128)
void gemm_kernel(const float* __restrict__ X1, const float* __restrict__ X2,
                 const float* __restrict__ W1, const float* __restrict__ b1,
                 const float* __restrict__ W2, const float* __restrict__ b2,
                 const float* __restrict__ Wlin, const float* __restrict__ blin,
                 float* __restrict__ out) {
    __shared__ float lds[16 * 65];      // padded: bank-conflict-free column reads

    int lane  = threadIdx.x & 31;
    int wave  = threadIdx.x >> 5;
    int row0  = blockIdx.x * 16;        // M tile base
    int ncol0 = wave * 16;              // N tile base
    int m     = lane & 15;              // A-row / B,C-column within tile
    int h     = lane >> 4;              // K/M half selector
    int ncol  = ncol0 + m;

    // ---- pass 1: X1@W1 + X2@W2 ----
    v8f c = {0.f, 0.f, 0.f, 0.f, 0.f, 0.f, 0.f, 0.f};
#pragma unroll
    for (int kk = 0; kk < 16; ++kk) {
        int k0 = kk * 4 + 2 * h;        // this lane-half's K offsets: k0, k0+1
        const float* x1r = X1 + (size_t)(row0 + m) * FIN + k0;
        const float* x2r = X2 + (size_t)(row0 + m) * FIN + k0;
        v2f a1 = { x1r[0], x1r[1] };
        v2f a2 = { x2r[0], x2r[1] };
        v2f bw1 = { W1[k0 * FIN + ncol], W1[(k0 + 1) * FIN + ncol] };
        v2f bw2 = { W2[k0 * FIN + ncol], W2[(k0 + 1) * FIN + ncol] };
        c = __builtin_amdgcn_wmma_f32_16x16x4_f32(false, a1, false, bw1,
                                                  (short)0, c, false, false);
        c = __builtin_amdgcn_wmma_f32_16x16x4_f32(false, a2, false, bw2,
                                                  (short)0, c, false, false);
    }
    float bias = b1[ncol] + b2[ncol];
#pragma unroll
    for (int r = 0; r < 8; ++r) c[r] += bias;

    // stage pre-activation tile in LDS (each wave owns its 16 columns)
#pragma unroll
    for (int r = 0; r < 8; ++r)
        lds[(r + 8 * h) * 65 + ncol] = c[r];
    __syncthreads();

    // ---- pass 2: relu(pre)@Wlin ----
    v8f c2 = {0.f, 0.f, 0.f, 0.f, 0.f, 0.f, 0.f, 0.f};
#pragma unroll
    for (int kk = 0; kk < 16; ++kk) {
        int k0 = kk * 4 + 2 * h;
        v2f a = { fmaxf(lds[m * 65 + k0], 0.0f),
                  fmaxf(lds[m * 65 + k0 + 1], 0.0f) };
        v2f b = { Wlin[k0 * FIN + ncol], Wlin[(k0 + 1) * FIN + ncol] };
        c2 = __builtin_amdgcn_wmma_f32_16x16x4_f32(false, a, false, b,
                                                   (short)0, c2, false, false);
    }
    float bl = blin[ncol];
#pragma unroll
    for (int r = 0; r < 8; ++r)
        out[(size_t)(row0 + r + 8 * h) * FIN + ncol] = c[r] + c2[r] + bl;
}

// ---------------- launch ----------------
extern "C" void kernel_launch(void* const* d_in, const int* in_sizes, int n_in,
                              void* d_out, int out_size, void* d_ws, size_t ws_size,
                              hipStream_t stream) {
    const float* u        = (const float*)d_in[0];
    const int*   coloring = (const int*)d_in[1];
    const int*   batch    = (const int*)d_in[2];
    const float* n_batch  = (const float*)d_in[3];
    // d_in[4] = num_graphs (scalar, known == 512)
    const float* W1   = (const float*)d_in[5];
    const float* b1   = (const float*)d_in[6];
    const float* W2   = (const float*)d_in[7];
    const float* b2   = (const float*)d_in[8];
    const float* Wlin = (const float*)d_in[9];
    const float* blin = (const float*)d_in[10];

    const int N = in_sizes[2];   // batch has shape (N,)

    float* ws       = (float*)d_ws;
    float* sum_diag = ws;                         // G*64
    float* sum_mean = ws + G_NUM * FIN;           // G*64
    float* cnt      = ws + 2 * G_NUM * FIN;       // G
    float* X1       = cnt + G_NUM;                // G*64
    float* X2       = X1 + G_NUM * FIN;           // G*64

    const int zcount = 2 * G_NUM * FIN + G_NUM;   // accumulators + counts
    zero_ws_kernel<<<(zcount + 255) / 256, 256, 0, stream>>>(ws, zcount);

    const int blocks  = 512;                      // 512*256 threads -> 8192 node streams
    const int columns = blocks * 256 / 16;
    accum_kernel<<<blocks, 256, 0, stream>>>(u, coloring, batch, n_batch,
                                             sum_diag, sum_mean, cnt, N, columns);

    finalize_kernel<<<(G_NUM * FIN + 255) / 256, 256, 0, stream>>>(sum_diag, sum_mean,
                                                                   cnt, X1, X2);

    gemm_kernel<<<G_NUM / 16, 128, 0, stream>>>(X1, X2, W1, b1, W2, b2,
                                                Wlin, blin, (float*)d_out);
}